// Gaussian3d_Full_77180562309430
// MI455X (gfx1250) — compile-verified
//
#include <hip/hip_runtime.h>

typedef __attribute__((ext_vector_type(2))) float v2f;
typedef __attribute__((ext_vector_type(8))) float v8f;

#define MDIM 64
#define GDIM 48
#define NPTS 4096
#define MG   (MDIM * GDIM)   // 3072 gaussian rows
#define KP   12              // K padded from 10 -> 12 (3 x K=4 WMMA steps)

// Raw hardware exp2: single v_exp_f32, no denorm-range fixup. Our exponent
// argument is always <= 0; anything below -126 is ~1e-38 and flushing it to
// zero is numerically irrelevant for the weighted sum.
#if __has_builtin(__builtin_amdgcn_exp2f)
#define FAST_EXP2(x) __builtin_amdgcn_exp2f(x)
#else
#define FAST_EXP2(x) exp2f(x)
#endif

// ---------------------------------------------------------------------------
// Kernel 1: per-(m,g) coefficient vector (10 coeffs, padded to 12), scaled by
// -0.5*log2(e) so the GEMM result feeds exp2 directly.
// quad(p) = p^T C p - 2 (C mu)^T p + mu^T C mu,  C = L L^T, L = R * diag(1/s)
// Stored transposed: CT[k*MG + mg] so 16 consecutive rows load coalesced.
// ---------------------------------------------------------------------------
__global__ __launch_bounds__(256) void g3d_coef_kernel(
    const float* __restrict__ mu_x, const float* __restrict__ mu_y,
    const float* __restrict__ mu_z, const float* __restrict__ sigmas,
    const float* __restrict__ r, float* __restrict__ CT) {
  int mg = blockIdx.x * blockDim.x + threadIdx.x;
  if (mg >= MG) return;

  float qw = r[mg * 4 + 0], qx = r[mg * 4 + 1];
  float qy = r[mg * 4 + 2], qz = r[mg * 4 + 3];
  float inv = rsqrtf(qw * qw + qx * qx + qy * qy + qz * qz);
  qw *= inv; qx *= inv; qy *= inv; qz *= inv;

  float R00 = 1.f - 2.f * (qy * qy + qz * qz);
  float R01 = 2.f * (qx * qy - qw * qz);
  float R02 = 2.f * (qx * qz + qw * qy);
  float R10 = 2.f * (qx * qy + qw * qz);
  float R11 = 1.f - 2.f * (qx * qx + qz * qz);
  float R12 = 2.f * (qy * qz - qw * qx);
  float R20 = 2.f * (qx * qz - qw * qy);
  float R21 = 2.f * (qy * qz + qw * qx);
  float R22 = 1.f - 2.f * (qx * qx + qy * qy);

  float is0 = 1.f / sigmas[mg * 3 + 0];
  float is1 = 1.f / sigmas[mg * 3 + 1];
  float is2 = 1.f / sigmas[mg * 3 + 2];

  float L00 = R00 * is0, L01 = R01 * is1, L02 = R02 * is2;
  float L10 = R10 * is0, L11 = R11 * is1, L12 = R12 * is2;
  float L20 = R20 * is0, L21 = R21 * is1, L22 = R22 * is2;

  float C00 = L00 * L00 + L01 * L01 + L02 * L02;
  float C01 = L00 * L10 + L01 * L11 + L02 * L12;
  float C02 = L00 * L20 + L01 * L21 + L02 * L22;
  float C11 = L10 * L10 + L11 * L11 + L12 * L12;
  float C12 = L10 * L20 + L11 * L21 + L12 * L22;
  float C22 = L20 * L20 + L21 * L21 + L22 * L22;

  float mx = mu_x[mg], my = mu_y[mg], mz = mu_z[mg];
  float lx = C00 * mx + C01 * my + C02 * mz;
  float ly = C01 * mx + C11 * my + C12 * mz;
  float lz = C02 * mx + C12 * my + C22 * mz;

  const float s = -0.5f * 1.4426950408889634f;  // fold -1/2 and log2(e)
  CT[0 * MG + mg]  = s * (mx * lx + my * ly + mz * lz);  // feature: 1
  CT[1 * MG + mg]  = s * (-2.f * lx);                    // x
  CT[2 * MG + mg]  = s * (-2.f * ly);                    // y
  CT[3 * MG + mg]  = s * (-2.f * lz);                    // z
  CT[4 * MG + mg]  = s * C00;                            // x^2
  CT[5 * MG + mg]  = s * C11;                            // y^2
  CT[6 * MG + mg]  = s * C22;                            // z^2
  CT[7 * MG + mg]  = s * (2.f * C01);                    // x*y
  CT[8 * MG + mg]  = s * (2.f * C02);                    // x*z
  CT[9 * MG + mg]  = s * (2.f * C12);                    // y*z
  CT[10 * MG + mg] = 0.f;                                // K padding
  CT[11 * MG + mg] = 0.f;
}

// ---------------------------------------------------------------------------
// Kernel 2: per-point feature vector F[k][n], k-major (12 x NPTS).
// Only y,z are rescaled (faithful to reference).
// ---------------------------------------------------------------------------
__global__ __launch_bounds__(256) void g3d_feat_kernel(
    const float* __restrict__ x, const float* __restrict__ y,
    const float* __restrict__ z, float* __restrict__ F) {
  int n = blockIdx.x * blockDim.x + threadIdx.x;
  if (n >= NPTS) return;
  float px = x[n];
  float py = (y[n] + 1.0f) * 0.5f * 2.0f;  // GRID_RANGE = 2
  float pz = (z[n] + 1.0f) * 0.5f * 2.0f;
  F[0 * NPTS + n]  = 1.0f;
  F[1 * NPTS + n]  = px;
  F[2 * NPTS + n]  = py;
  F[3 * NPTS + n]  = pz;
  F[4 * NPTS + n]  = px * px;
  F[5 * NPTS + n]  = py * py;
  F[6 * NPTS + n]  = pz * pz;
  F[7 * NPTS + n]  = px * py;
  F[8 * NPTS + n]  = px * pz;
  F[9 * NPTS + n]  = py * pz;
  F[10 * NPTS + n] = 0.f;
  F[11 * NPTS + n] = 0.f;
}

// ---------------------------------------------------------------------------
// Kernel 3: fused WMMA GEMM (3072x12 @ 12x4096) + exp2 + weighted reduce.
// Block = 256 threads = 8 waves. blockIdx.y = m (feature row of output).
// Each wave: one 16-column tile of points, 3 row tiles (48 gaussians of m),
// K=12 via 3 chained V_WMMA_F32_16X16X4_F32.
//
// Fragment layouts (CDNA5 ISA 7.12.2, f32 16x16x4):
//   A (16x4): lane L -> M = L%16 ; VGPR0 = K(2*(L/16)), VGPR1 = K(2*(L/16)+1)
//   B (4x16): lane L -> N = L%16 ; VGPR0 = K(2*(L/16)), VGPR1 = K(2*(L/16)+1)
//   C/D:      lane L -> N = L%16 ; VGPR v -> M = v + 8*(L/16)
// No divergence before WMMA (EXEC all ones requirement).
// ---------------------------------------------------------------------------
__global__ __launch_bounds__(256) void g3d_gmm_kernel(
    const float* __restrict__ CT, const float* __restrict__ F,
    const float* __restrict__ weight, float* __restrict__ out) {
  int lane = threadIdx.x & 31;
  int wave = threadIdx.x >> 5;
  int half = lane >> 4;       // 0: lanes 0-15, 1: lanes 16-31
  int lm   = lane & 15;
  int m    = blockIdx.y;
  int nb   = (blockIdx.x * 8 + wave) * 16;   // column tile base
  int col  = nb + lm;

  // B fragments for the 3 K-chunks (shared by all 3 row tiles)
  v2f b[3];
#pragma unroll
  for (int kc = 0; kc < 3; ++kc) {
    int k0 = kc * 4 + 2 * half;
    b[kc].x = F[k0 * NPTS + col];
    b[kc].y = F[(k0 + 1) * NPTS + col];
  }

  float partial = 0.f;
#pragma unroll
  for (int rt = 0; rt < 3; ++rt) {
    int rowbase = m * GDIM + rt * 16;
    v8f acc = {};
#pragma unroll
    for (int kc = 0; kc < 3; ++kc) {
      int k0 = kc * 4 + 2 * half;
      v2f a;
      a.x = CT[k0 * MG + rowbase + lm];
      a.y = CT[(k0 + 1) * MG + rowbase + lm];
      acc = __builtin_amdgcn_wmma_f32_16x16x4_f32(
          /*neg_a=*/false, a, /*neg_b=*/false, b[kc],
          /*c_mod=*/(short)0, acc, /*reuse_a=*/false, /*reuse_b=*/false);
    }
    // epilogue: pdf = exp2(acc) (log2e folded in), weighted G-reduction.
    // Raw v_exp_f32: argument is always <= 0; sub-denormal results flush to 0.
#pragma unroll
    for (int v = 0; v < 8; ++v) {
      int g = rt * 16 + v + 8 * half;
      partial += weight[m * GDIM + g] * FAST_EXP2(acc[v]);
    }
  }

  // combine the two lane halves (rows 0-7 vs 8-15 of each row tile)
  partial += __shfl_xor(partial, 16, 32);
  if (lane < 16) out[col * MDIM + m] = partial;  // out is (N, M) row-major
}

// ---------------------------------------------------------------------------
extern "C" void kernel_launch(void* const* d_in, const int* in_sizes, int n_in,
                              void* d_out, int out_size, void* d_ws, size_t ws_size,
                              hipStream_t stream) {
  const float* x      = (const float*)d_in[0];
  const float* y      = (const float*)d_in[1];
  const float* z      = (const float*)d_in[2];
  const float* mu_x   = (const float*)d_in[3];
  const float* mu_y   = (const float*)d_in[4];
  const float* mu_z   = (const float*)d_in[5];
  const float* sigmas = (const float*)d_in[6];
  const float* r      = (const float*)d_in[7];
  const float* weight = (const float*)d_in[8];
  float* out = (float*)d_out;

  float* CT = (float*)d_ws;        // KP * MG   floats = 147456 B
  float* F  = CT + KP * MG;        // KP * NPTS floats = 196608 B

  g3d_coef_kernel<<<(MG + 255) / 256, 256, 0, stream>>>(mu_x, mu_y, mu_z,
                                                        sigmas, r, CT);
  g3d_feat_kernel<<<(NPTS + 255) / 256, 256, 0, stream>>>(x, y, z, F);

  dim3 grid(NPTS / 128, MDIM);  // 8 waves/block * 16 cols = 128 cols per block
  g3d_gmm_kernel<<<grid, 256, 0, stream>>>(CT, F, weight, out);
}